// GlobalAttention_4707284157172
// MI455X (gfx1250) — compile-verified
//
#include <hip/hip_runtime.h>
#include <math.h>

// ---- problem constants ----
#define HEADS   8
#define DHEAD   32
#define DMODEL  256
#define DM2     512

typedef __attribute__((ext_vector_type(2))) float v2f;
typedef __attribute__((ext_vector_type(8))) float v8f;

__device__ __forceinline__ unsigned f2ord(float f) {
  unsigned u = __float_as_uint(f);
  return (u >> 31) ? ~u : (u | 0x80000000u);
}
__device__ __forceinline__ float ord2f(unsigned s) {
  unsigned u = (s >> 31) ? (s & 0x7FFFFFFFu) : ~s;
  return __uint_as_float(u);
}

// =====================================================================
// fp32 WMMA GEMM: C = A(MxK) @ W(KxN) + bias
// One wave computes a 32x64 tile of C (2 M-subtiles x 4 N-subtiles),
// 8x V_WMMA_F32_16X16X4_F32 per K-step of 4; A/B fragments reused in regs.
// =====================================================================
__global__ void gemm_wmma_f32(const float* __restrict__ A, const float* __restrict__ W,
                              const float* __restrict__ bias, float* __restrict__ C,
                              int M, int K, int Ncols) {
  const int wavesPerBlock = blockDim.x >> 5;
  const int wave = blockIdx.x * wavesPerBlock + (threadIdx.x >> 5);
  const int tilesN = Ncols >> 6;                    // 64-wide N tiles
  const int tilesM = (M + 31) >> 5;                 // 32-tall M tiles
  if (wave >= tilesM * tilesN) return;              // uniform per-wave exit
  const int tm = wave / tilesN;
  const int tn = wave - tm * tilesN;
  const int row0 = tm << 5, col0 = tn << 6;
  const int lane = threadIdx.x & 31;
  const int half = lane >> 4;                       // 0: K=0,1  1: K=2,3
  const int l    = lane & 15;

  const int ra0 = min(row0 + l,      M - 1);        // clamped A rows (OOB-safe)
  const int ra1 = min(row0 + 16 + l, M - 1);
  const float* __restrict__ pA0 = A + (size_t)ra0 * K + (half << 1);
  const float* __restrict__ pA1 = A + (size_t)ra1 * K + (half << 1);
  const float* __restrict__ pB  = W + (size_t)(half << 1) * Ncols + col0 + l;
  const size_t bstep = (size_t)4 * Ncols;

  v8f acc0[4] = {}, acc1[4] = {};
  for (int k0 = 0; k0 < K; k0 += 4) {
    __builtin_prefetch(pB + 2 * bstep, 0, 1);       // strided B stream -> global_prefetch_b8
    v2f a0, a1;
    a0.x = pA0[0]; a0.y = pA0[1];
    a1.x = pA1[0]; a1.y = pA1[1];
#pragma unroll
    for (int j = 0; j < 4; ++j) {
      v2f b;
      b.x = pB[j * 16];
      b.y = pB[j * 16 + Ncols];
      acc0[j] = __builtin_amdgcn_wmma_f32_16x16x4_f32(false, a0, false, b,
                                                      (short)0, acc0[j], false, false);
      acc1[j] = __builtin_amdgcn_wmma_f32_16x16x4_f32(false, a1, false, b,
                                                      (short)0, acc1[j], false, false);
    }
    pA0 += 4; pA1 += 4; pB += bstep;
  }

#pragma unroll
  for (int j = 0; j < 4; ++j) {
    const int col = col0 + (j << 4) + l;
    const float bv = bias[col];
#pragma unroll
    for (int r = 0; r < 8; ++r) {
      const int rowA = row0 + r + (half << 3);
      const int rowB = rowA + 16;
      if (rowA < M) C[(size_t)rowA * Ncols + col] = acc0[j][r] + bv;
      if (rowB < M) C[(size_t)rowB * Ncols + col] = acc1[j][r] + bv;
    }
  }
}

// GEMM with gathered concat A: A[row][k] = k<256 ? A0[row][k] : A1[gidx[row]][k-256]; K=512.
__global__ void gemm_wmma_cat_gather(const float* __restrict__ A0, const float* __restrict__ A1,
                                     const int* __restrict__ gidx,
                                     const float* __restrict__ W, const float* __restrict__ bias,
                                     float* __restrict__ C, int M, int Ncols) {
  const int wavesPerBlock = blockDim.x >> 5;
  const int wave = blockIdx.x * wavesPerBlock + (threadIdx.x >> 5);
  const int tilesN = Ncols >> 6;
  const int tilesM = (M + 31) >> 5;
  if (wave >= tilesM * tilesN) return;
  const int tm = wave / tilesN;
  const int tn = wave - tm * tilesN;
  const int row0 = tm << 5, col0 = tn << 6;
  const int lane = threadIdx.x & 31;
  const int half = lane >> 4;
  const int l    = lane & 15;

  const int ra0 = min(row0 + l,      M - 1);
  const int ra1 = min(row0 + 16 + l, M - 1);
  const float* __restrict__ F0 = A0 + (size_t)ra0 * DMODEL + (half << 1);       // f_bond rows
  const float* __restrict__ F1 = A0 + (size_t)ra1 * DMODEL + (half << 1);
  const float* __restrict__ G0 = A1 + (size_t)gidx[ra0] * DMODEL + (half << 1); // out[src] rows
  const float* __restrict__ G1 = A1 + (size_t)gidx[ra1] * DMODEL + (half << 1);
  const float* __restrict__ pB = W + (size_t)(half << 1) * Ncols + col0 + l;
  const size_t bstep = (size_t)4 * Ncols;

  v8f acc0[4] = {}, acc1[4] = {};
  for (int k0 = 0; k0 < DM2; k0 += 4) {
    __builtin_prefetch(pB + 2 * bstep, 0, 1);
    v2f a0, a1;
    if (k0 < DMODEL) {                 // ka = k0 + 2*half stays < 256 (k0 <= 252)
      a0.x = F0[k0]; a0.y = F0[k0 + 1];
      a1.x = F1[k0]; a1.y = F1[k0 + 1];
    } else {
      const int kg = k0 - DMODEL;
      a0.x = G0[kg]; a0.y = G0[kg + 1];
      a1.x = G1[kg]; a1.y = G1[kg + 1];
    }
#pragma unroll
    for (int j = 0; j < 4; ++j) {
      v2f b;
      b.x = pB[j * 16];
      b.y = pB[j * 16 + Ncols];
      acc0[j] = __builtin_amdgcn_wmma_f32_16x16x4_f32(false, a0, false, b,
                                                      (short)0, acc0[j], false, false);
      acc1[j] = __builtin_amdgcn_wmma_f32_16x16x4_f32(false, a1, false, b,
                                                      (short)0, acc1[j], false, false);
    }
    pB += bstep;
  }

#pragma unroll
  for (int j = 0; j < 4; ++j) {
    const int col = col0 + (j << 4) + l;
    const float bv = bias[col];
#pragma unroll
    for (int r = 0; r < 8; ++r) {
      const int rowA = row0 + r + (half << 3);
      const int rowB = rowA + 16;
      if (rowA < M) C[(size_t)rowA * Ncols + col] = acc0[j][r] + bv;
      if (rowB < M) C[(size_t)rowB * Ncols + col] = acc1[j][r] + bv;
    }
  }
}

// ===================== elementwise / attention kernels =====================
__global__ void zero_f32(float* __restrict__ p, long n) {
  long i = (long)blockIdx.x * blockDim.x + threadIdx.x;
  if (i < n) p[i] = 0.0f;
}

// kg = k_global + deg_emb[deg]  (in-place on cols 256..511 of k)
__global__ void add_deg(float* __restrict__ kbuf, const int* __restrict__ deg,
                        const float* __restrict__ deg_emb, int N) {
  int i = blockIdx.x * blockDim.x + threadIdx.x;
  if (i >= N * DMODEL) return;
  int n = i >> 8, c = i & 255;
  kbuf[(size_t)n * DM2 + DMODEL + c] += deg_emb[(size_t)deg[n] * DMODEL + c];
}

#define SCALE 0.17677669529663687f  // 1/sqrt(32)

__global__ void local_score(const float* __restrict__ q, const float* __restrict__ k,
                            const float* __restrict__ abvb,
                            const int* __restrict__ src, const int* __restrict__ dst,
                            float* __restrict__ score, unsigned* __restrict__ nmax,
                            int E, int EB) {
  int i = blockIdx.x * blockDim.x + threadIdx.x;
  if (i >= E * HEADS) return;
  int e = i >> 3, h = i & 7;
  int s = src[e], d = dst[e];
  const float* qp = q + (size_t)d * DM2 + h * DHEAD;     // q_local (cols 0..255)
  const float* kp = k + (size_t)s * DM2 + h * DHEAD;     // k_local
  float acc = 0.0f;
  if (e < EB) {
    const float* ab = abvb + (size_t)e * DM2 + h * DHEAD;  // attention_bond half
    for (int t = 0; t < DHEAD; ++t) acc += qp[t] * kp[t] + ab[t];
  } else {
    for (int t = 0; t < DHEAD; ++t) acc += qp[t] * kp[t];
  }
  acc *= SCALE;
  score[i] = acc;
  atomicMax(&nmax[d * HEADS + h], f2ord(acc));
}

__global__ void seg_exp(float* __restrict__ score, const int* __restrict__ seg,
                        const unsigned* __restrict__ nmax, float* __restrict__ nsum, int E) {
  int i = blockIdx.x * blockDim.x + threadIdx.x;
  if (i >= E * HEADS) return;
  int e = i >> 3, h = i & 7;
  int sgi = seg[e] * HEADS + h;
  float ex = expf(score[i] - ord2f(nmax[sgi]));
  score[i] = ex;
  atomicAdd(&nsum[sgi], ex);
}

__global__ void local_accum(const float* __restrict__ score, const float* __restrict__ nsum,
                            const float* __restrict__ v, const float* __restrict__ abvb,
                            const int* __restrict__ src, const int* __restrict__ dst,
                            float* __restrict__ cat, int E, int EB) {
  int i = blockIdx.x * blockDim.x + threadIdx.x;
  if (i >= E * DMODEL) return;
  int e = i >> 8, c = i & 255, h = c >> 5;
  int s = src[e], d = dst[e];
  float w = score[e * HEADS + h] / nsum[d * HEADS + h];
  float val = v[(size_t)s * DM2 + c];                       // v_local
  if (e < EB) val += abvb[(size_t)e * DM2 + DMODEL + c];    // value_bond half
  atomicAdd(&cat[(size_t)d * DM2 + c], w * val);            // local_out: cols 0..255
}

__global__ void global_score(const float* __restrict__ q, const float* __restrict__ kg,
                             const float* __restrict__ dist_emb,
                             const int* __restrict__ qi, const int* __restrict__ ki,
                             const int* __restrict__ distm,
                             float* __restrict__ score, int E) {
  int i = blockIdx.x * blockDim.x + threadIdx.x;
  if (i >= E * HEADS) return;
  int e = i >> 3, h = i & 7;
  const float* qp = q + (size_t)qi[e] * DM2 + DMODEL + h * DHEAD;   // q_global
  const float* kp = kg + (size_t)ki[e] * DM2 + DMODEL + h * DHEAD;  // kg (k_global + deg)
  const float* dp = dist_emb + (size_t)distm[e] * DMODEL + h * DHEAD;
  float acc = 0.0f;
  for (int t = 0; t < DHEAD; ++t) acc += qp[t] * kp[t] + dp[t];
  score[i] = acc * SCALE;
}

__global__ void global_ab_scatter(const float* __restrict__ abvb, const int* __restrict__ abi,
                                  float* __restrict__ score, int EB) {
  int i = blockIdx.x * blockDim.x + threadIdx.x;
  if (i >= EB * HEADS) return;
  int j = i >> 3, h = i & 7;
  const float* ab = abvb + (size_t)j * DM2 + h * DHEAD;
  float s = 0.0f;
  for (int t = 0; t < DHEAD; ++t) s += ab[t];
  atomicAdd(&score[(size_t)abi[j] * HEADS + h], s * SCALE);
}

__global__ void global_max(const float* __restrict__ score, const int* __restrict__ qi,
                           unsigned* __restrict__ nmax, int E) {
  int i = blockIdx.x * blockDim.x + threadIdx.x;
  if (i >= E * HEADS) return;
  int e = i >> 3, h = i & 7;
  atomicMax(&nmax[qi[e] * HEADS + h], f2ord(score[i]));
}

__global__ void global_accum(const float* __restrict__ score, const float* __restrict__ nsum,
                             const float* __restrict__ v,
                             const int* __restrict__ qi, const int* __restrict__ ki,
                             float* __restrict__ cat, int E) {
  int i = blockIdx.x * blockDim.x + threadIdx.x;
  if (i >= E * DMODEL) return;
  int e = i >> 8, c = i & 255, h = c >> 5;
  float w = score[e * HEADS + h] / nsum[qi[e] * HEADS + h];
  float val = v[(size_t)ki[e] * DM2 + DMODEL + c];          // v_global
  atomicAdd(&cat[(size_t)qi[e] * DM2 + DMODEL + c], w * val); // global_out: cols 256..511
}

// ===================== launch =====================
static inline int cdiv(long a, long b) { return (int)((a + b - 1) / b); }

extern "C" void kernel_launch(void* const* d_in, const int* in_sizes, int n_in,
                              void* d_out, int out_size, void* d_ws, size_t ws_size,
                              hipStream_t stream) {
  const float* f_node   = (const float*)d_in[0];
  const float* f_bond   = (const float*)d_in[1];
  const int*   deg      = (const int*)d_in[2];
  const int*   distm    = (const int*)d_in[3];
  const int*   bidx     = (const int*)d_in[4];
  const int*   qidx     = (const int*)d_in[5];
  const int*   kidx     = (const int*)d_in[6];
  const int*   abi      = (const int*)d_in[7];
  const float* deg_emb  = (const float*)d_in[8];
  const float* dist_emb = (const float*)d_in[9];
  const float* Wq_w = (const float*)d_in[10]; const float* Wq_b = (const float*)d_in[11];
  const float* Wk_w = (const float*)d_in[12]; const float* Wk_b = (const float*)d_in[13];
  const float* Wv_w = (const float*)d_in[14]; const float* Wv_b = (const float*)d_in[15];
  const float* out_w = (const float*)d_in[16]; const float* out_b = (const float*)d_in[17];
  const float* be_w  = (const float*)d_in[18]; const float* be_b  = (const float*)d_in[19];
  const float* bu_w  = (const float*)d_in[20]; const float* bu_b  = (const float*)d_in[21];

  const int N  = in_sizes[0] / DMODEL;   // 20000
  const int EB = in_sizes[1] / DMODEL;   // 60000
  const int EL = in_sizes[4] / 2;        // 80000
  const int EG = in_sizes[5];            // 400000
  const int* src = bidx;
  const int* dst = bidx + EL;

  // ---- workspace layout (floats) ----
  float* ws = (float*)d_ws;
  size_t off = 0;
  float* q    = ws + off; off += (size_t)N * DM2;
  float* k    = ws + off; off += (size_t)N * DM2;
  float* v    = ws + off; off += (size_t)N * DM2;
  float* abvb = ws + off; off += (size_t)EB * DM2;
  float* cat  = ws + off; off += (size_t)N * DM2;     // [local_out | global_out]
  unsigned* nmaxL = (unsigned*)(ws + off); off += (size_t)N * HEADS;
  float*    nsumL = ws + off;              off += (size_t)N * HEADS;
  unsigned* nmaxG = (unsigned*)(ws + off); off += (size_t)N * HEADS;
  float*    nsumG = ws + off;              off += (size_t)N * HEADS;
  float* scoreL = ws + off; off += (size_t)EL * HEADS;
  float* scoreG = ws + off; off += (size_t)EG * HEADS;
  float* out = (float*)d_out;               // (N,256)
  float* bond_out = out + (size_t)N * DMODEL;

  const int TB = 256;      // elementwise block
  const int GB = 128;      // 4 waves per block for GEMM

  // zero the accumulator region: cat + 4 stats arrays (contiguous; encoded -inf == 0 bits)
  long nz = (long)N * DM2 + 4L * N * HEADS;
  zero_f32<<<cdiv(nz, TB), TB, 0, stream>>>(cat, nz);

  // GEMM: ab_vb = f_bond @ bond_emb_w + b   (EBx256 @ 256x512)
  gemm_wmma_f32<<<cdiv((long)((EB + 31) / 32) * (DM2 / 64), 4), GB, 0, stream>>>(
      f_bond, be_w, be_b, abvb, EB, DMODEL, DM2);
  // GEMMs: q,k,v = f_node @ W + b   (Nx256 @ 256x512)
  gemm_wmma_f32<<<cdiv((long)((N + 31) / 32) * (DM2 / 64), 4), GB, 0, stream>>>(
      f_node, Wq_w, Wq_b, q, N, DMODEL, DM2);
  gemm_wmma_f32<<<cdiv((long)((N + 31) / 32) * (DM2 / 64), 4), GB, 0, stream>>>(
      f_node, Wk_w, Wk_b, k, N, DMODEL, DM2);
  gemm_wmma_f32<<<cdiv((long)((N + 31) / 32) * (DM2 / 64), 4), GB, 0, stream>>>(
      f_node, Wv_w, Wv_b, v, N, DMODEL, DM2);

  // kg = k_global + deg_emb[deg]
  add_deg<<<cdiv((long)N * DMODEL, TB), TB, 0, stream>>>(k, deg, deg_emb, N);

  // ---- local attention ----
  local_score<<<cdiv((long)EL * HEADS, TB), TB, 0, stream>>>(q, k, abvb, src, dst,
                                                             scoreL, nmaxL, EL, EB);
  seg_exp<<<cdiv((long)EL * HEADS, TB), TB, 0, stream>>>(scoreL, dst, nmaxL, nsumL, EL);
  local_accum<<<cdiv((long)EL * DMODEL, TB), TB, 0, stream>>>(scoreL, nsumL, v, abvb,
                                                              src, dst, cat, EL, EB);

  // ---- global attention ----
  global_score<<<cdiv((long)EG * HEADS, TB), TB, 0, stream>>>(q, k, dist_emb, qidx, kidx,
                                                              distm, scoreG, EG);
  global_ab_scatter<<<cdiv((long)EB * HEADS, TB), TB, 0, stream>>>(abvb, abi, scoreG, EB);
  global_max<<<cdiv((long)EG * HEADS, TB), TB, 0, stream>>>(scoreG, qidx, nmaxG, EG);
  seg_exp<<<cdiv((long)EG * HEADS, TB), TB, 0, stream>>>(scoreG, qidx, nmaxG, nsumG, EG);
  global_accum<<<cdiv((long)EG * DMODEL, TB), TB, 0, stream>>>(scoreG, nsumG, v, qidx,
                                                               kidx, cat, EG);

  // out = cat @ out_w + out_b   (Nx512 @ 512x256)
  gemm_wmma_f32<<<cdiv((long)((N + 31) / 32) * (DMODEL / 64), 4), GB, 0, stream>>>(
      cat, out_w, out_b, out, N, DM2, DMODEL);

  // bond_out = [f_bond | out[src]] @ bond_update_w + b  (EBx512 @ 512x256)
  gemm_wmma_cat_gather<<<cdiv((long)((EB + 31) / 32) * (DMODEL / 64), 4), GB, 0, stream>>>(
      f_bond, out, src, bu_w, bu_b, bond_out, EB, DMODEL);
}